// MCAttEGNN_9199819948217
// MI455X (gfx1250) — compile-verified
//
#include <hip/hip_runtime.h>
#include <hip/hip_bf16.h>
#include <math.h>

#define NN   10000
#define HH   128
#define INNF 64
#define ONF  64
#define ECTX 160000
#define EATT 80000

typedef float v2f __attribute__((ext_vector_type(2)));
typedef float v8f __attribute__((ext_vector_type(8)));

__device__ inline float fast_silu(float v) {
  // v * sigmoid(v); v_exp_f32 + v_rcp_f32 (avoids IEEE div expansion)
  return v * __builtin_amdgcn_rcpf(1.0f + __expf(-v));
}

// ---------------------------------------------------------------------------
// Generic gather-GEMM with CDNA5 f32 WMMA (V_WMMA_F32_16X16X4_F32).
// Logical A row r = [ A0[map0[r]] (K0 floats) | A1[map1[r]] (K1) | A2[r] (K2) ]
// out[M x N] = act( A @ W[K x N] + bias ) (+ resid).  flags: 1=SiLU, 2=resid.
//
// Block = 128 threads (4 waves), tile = 16(M) x 64(N); wave w owns cols
// n0 = blockIdx.y*64 + w*16.  The 16 x Ktot logical A tile (all concat
// segments, rows gathered via map) is staged into LDS once per block, so the
// WMMA K-loop is branch-free and EXEC stays all-ones (WMMA requirement).
// Requires M mult of 16, N mult of 64; Ktot mult of 16 (true for 64/128/144/256/272).
//
// A-frag (ISA 7.12.2): lanes 0-15 (M=row) hold {K,K+1}; lanes 16-31 {K+2,K+3}.
// B-frag: VGPR0 = {row K | row K+2}, VGPR1 = {row K+1 | row K+3}, col=lane%16.
// C/D: VGPR i -> row m0+i (lanes 0-15) / m0+i+8 (lanes 16-31), col = lane%16.
// ---------------------------------------------------------------------------
#define KMAX 272
__global__ void gemm_wmma_f32(
    const float* __restrict__ A0, const int* __restrict__ map0, int K0,
    const float* __restrict__ A1, const int* __restrict__ map1, int K1,
    const float* __restrict__ A2, int K2,
    const float* __restrict__ W,  const float* __restrict__ bias,
    const float* __restrict__ resid, float* __restrict__ out,
    int M, int Ktot, int N, int flags)
{
  __shared__ __align__(16) float As[16 * (KMAX + 2)];
  const int tid  = threadIdx.x;
  const int lane = tid & 31;
  const int wave = tid >> 5;
  const int half = lane >> 4;
  const int l16  = lane & 15;
  const int m0 = blockIdx.x << 4;
  const int n0 = (blockIdx.y << 6) + (wave << 4);
  const int stride = Ktot + 2;                 // even; breaks bank alignment

  // ---- stage logical A tile (16 rows x Ktot) into LDS ----
  for (int rr = 0; rr < 16; ++rr) {
    const int m = m0 + rr;
    float* dst = As + rr * stride;
    const float* s0 = A0 + (size_t)(map0 ? map0[m] : m) * K0;
    for (int k = tid; k < K0; k += 128) dst[k] = s0[k];
    if (K1 > 0) {
      const float* s1 = A1 + (size_t)(map1 ? map1[m] : m) * K1;
      for (int k = tid; k < K1; k += 128) dst[K0 + k] = s1[k];
    }
    if (K2 > 0) {
      const float* s2 = A2 + (size_t)m * K2;
      for (int k = tid; k < K2; k += 128) dst[K0 + K1 + k] = s2[k];
    }
  }
  __syncthreads();

  // ---- WMMA K-loop, 4 wmma per outer iteration, branch-free ----
  const float* arow = As + l16 * stride;       // this lane's A row (LDS)
  const float* wcol = W + n0 + l16;            // this lane's W column
  v8f c = {0.f,0.f,0.f,0.f,0.f,0.f,0.f,0.f};
  for (int k0 = 0; k0 < Ktot; k0 += 16) {
    #pragma unroll
    for (int u = 0; u < 4; ++u) {
      const int kk = k0 + (u << 2) + (half << 1);
      v2f a = *(const v2f*)(arow + kk);        // ds_load_b64
      const float* wp = wcol + (size_t)kk * N;
      v2f b; b.x = wp[0]; b.y = wp[N];
      c = __builtin_amdgcn_wmma_f32_16x16x4_f32(false, a, false, b,
                                                (short)0, c, false, false);
    }
  }

  // ---- epilogue: bias hoisted (n is invariant across the 8 rows) ----
  const int n = n0 + l16;
  const float bv = bias ? bias[n] : 0.f;
  #pragma unroll
  for (int i = 0; i < 8; ++i) {
    const int m = m0 + i + (half << 3);
    float v = c[i] + bv;
    if (flags & 1) v = fast_silu(v);
    if (flags & 2) v += resid[(size_t)m * N + n];
    out[(size_t)m * N + n] = v;
  }
}

// ------------------------------- helpers -----------------------------------
__global__ void k_copy(float* __restrict__ d, const float* __restrict__ s, int n){
  int i = blockIdx.x*blockDim.x + threadIdx.x; if (i < n) d[i] = s[i];
}
__global__ void k_add(float* __restrict__ d, const float* __restrict__ s, int n){
  int i = blockIdx.x*blockDim.x + threadIdx.x; if (i < n) d[i] += s[i];
}

// cd[e] = x[row]-x[col] (4x3); radial[e,a,b] = <cd[a],cd[b]>; sumsq += radial^2
__global__ void k_cd_radial(const float* __restrict__ x, const int* __restrict__ row,
                            const int* __restrict__ col, float* __restrict__ cd,
                            float* __restrict__ radial, float* __restrict__ sumsq, int E)
{
  __shared__ float ssq[16];
  if (threadIdx.x < 16) ssq[threadIdx.x] = 0.f;
  __syncthreads();
  int e = blockIdx.x*blockDim.x + threadIdx.x;
  if (e < E) {
    int r = row[e], q = col[e];
    float d[12];
    #pragma unroll
    for (int k = 0; k < 12; ++k) d[k] = x[r*12+k] - x[q*12+k];
    #pragma unroll
    for (int k = 0; k < 12; ++k) cd[(size_t)e*12+k] = d[k];
    #pragma unroll
    for (int a = 0; a < 4; ++a)
      #pragma unroll
      for (int b = 0; b < 4; ++b) {
        float v = d[a*3]*d[b*3] + d[a*3+1]*d[b*3+1] + d[a*3+2]*d[b*3+2];
        radial[(size_t)e*16 + a*4 + b] = v;
        atomicAdd(&ssq[a*4+b], v*v);
      }
  }
  __syncthreads();
  if (threadIdx.x < 16) atomicAdd(&sumsq[threadIdx.x], ssq[threadIdx.x]);
}

__global__ void k_radial_norm(float* __restrict__ radial,
                              const float* __restrict__ sumsq, int total){
  int i = blockIdx.x*blockDim.x + threadIdx.x;
  if (i < total) radial[i] /= fmaxf(sqrtf(sumsq[i & 15]), 1e-12f);
}

// cw[e,c] = dot(t[e,:128], W[:,c])  (coord_mlp2, 128x4, no bias)
__global__ void k_matvec4(const float* __restrict__ t, const float* __restrict__ W,
                          float* __restrict__ cw, int E){
  __shared__ float w[512];
  for (int i = threadIdx.x; i < 512; i += blockDim.x) w[i] = W[i];
  __syncthreads();
  int idx = blockIdx.x*blockDim.x + threadIdx.x;
  if (idx < E*4) {
    int e = idx >> 2, c = idx & 3;
    const float* tp = t + (size_t)e*128;
    float acc = 0.f;
    for (int j = 0; j < 128; ++j) acc += tp[j]*w[j*4+c];
    cw[idx] = acc;
  }
}

__global__ void k_coord_scatter(const int* __restrict__ row, const float* __restrict__ cd,
                                const float* __restrict__ cw, float* __restrict__ snode,
                                float* __restrict__ cnt, int E){
  int e = blockIdx.x*blockDim.x + threadIdx.x;
  if (e < E) {
    int r = row[e];
    #pragma unroll
    for (int k = 0; k < 12; ++k)
      atomicAdd(&snode[r*12+k], cd[(size_t)e*12+k]*cw[(size_t)e*4 + k/3]);
    atomicAdd(&cnt[r], 1.0f);
  }
}
__global__ void k_coord_mean(float* __restrict__ x, const float* __restrict__ s,
                             const float* __restrict__ cnt, int total){
  int i = blockIdx.x*blockDim.x + threadIdx.x;
  if (i < total) x[i] += s[i]/fmaxf(cnt[i/12], 1.0f);
}
__global__ void k_agg_rows(const int* __restrict__ row, const float* __restrict__ ef,
                           float* __restrict__ agg, int E){
  int idx = blockIdx.x*blockDim.x + threadIdx.x;
  if (idx < E*128) { int e = idx >> 7, j = idx & 127;
    atomicAdd(&agg[row[e]*128 + j], ef[idx]); }
}

// ------- attention: ordered-uint trick for float atomic segment max --------
__device__ inline unsigned f2ord(float f){ unsigned u = __float_as_uint(f);
  return (u & 0x80000000u) ? ~u : (u | 0x80000000u); }
__device__ inline float ord2f(unsigned u){
  return (u & 0x80000000u) ? __uint_as_float(u & 0x7fffffffu) : __uint_as_float(~u); }

__global__ void k_logit_max(const int* __restrict__ row, const float* __restrict__ q,
                            const float* __restrict__ kv, float* __restrict__ logit,
                            unsigned* __restrict__ mbuf, int E){
  int e = blockIdx.x*blockDim.x + threadIdx.x;
  if (e < E) {
    const float* qp = q + (size_t)e*128;
    const float* kp = kv + (size_t)e*256;
    float acc = 0.f;
    for (int j = 0; j < 128; ++j) acc += qp[j]*kp[2*j];
    logit[e] = acc;
    atomicMax(&mbuf[row[e]], f2ord(acc));
  }
}
__global__ void k_exp_sum(const int* __restrict__ row, float* __restrict__ logit,
                          const unsigned* __restrict__ mbuf, float* __restrict__ sbuf, int E){
  int e = blockIdx.x*blockDim.x + threadIdx.x;
  if (e < E) { float m = ord2f(mbuf[row[e]]); float ex = __expf(logit[e]-m);
    logit[e] = ex; atomicAdd(&sbuf[row[e]], ex); }
}
__global__ void k_alpha_div(const int* __restrict__ row, float* __restrict__ a,
                            const float* __restrict__ sbuf, int E){
  int e = blockIdx.x*blockDim.x + threadIdx.x;
  if (e < E) a[e] /= sbuf[row[e]];
}
__global__ void k_hagg(const int* __restrict__ row, const float* __restrict__ alpha,
                       const float* __restrict__ kv, float* __restrict__ agg, int E){
  int idx = blockIdx.x*blockDim.x + threadIdx.x;
  if (idx < E*128) { int e = idx >> 7, j = idx & 127;
    atomicAdd(&agg[row[e]*128 + j], alpha[e]*kv[(size_t)e*256 + 2*j + 1]); }
}
__global__ void k_extract_v(const float* __restrict__ kv, float* __restrict__ V, int E){
  int idx = blockIdx.x*blockDim.x + threadIdx.x;
  if (idx < E*128) { int e = idx >> 7, j = idx & 127;
    V[idx] = kv[(size_t)e*256 + 2*j + 1]; }
}
__global__ void k_att_coord_scatter(const int* __restrict__ row, const float* __restrict__ cd,
                                    const float* __restrict__ cw, const float* __restrict__ alpha,
                                    float* __restrict__ snode, int E){
  int e = blockIdx.x*blockDim.x + threadIdx.x;
  if (e < E) {
    int r = row[e]; float al = alpha[e];
    #pragma unroll
    for (int k = 0; k < 12; ++k)
      atomicAdd(&snode[r*12+k], cd[(size_t)e*12+k]*cw[(size_t)e*4 + k/3]*al);
  }
}

// ---------------------------------------------------------------------------
extern "C" void kernel_launch(void* const* d_in, const int* in_sizes, int n_in,
                              void* d_out, int out_size, void* d_ws, size_t ws_size,
                              hipStream_t stream)
{
  if (n_in < 73) return;
  const float* h_in = (const float*)d_in[0];
  const float* x_in = (const float*)d_in[1];
  const int* ctx_row = (const int*)d_in[2]; const int* ctx_col = ctx_row + ECTX;
  const int* att_row = (const int*)d_in[3]; const int* att_col = att_row + EATT;

  // params flattened in JAX sorted-key pytree order:
  // att[0..2]{coord_mlp1.W,b, coord_mlp2.W, kv.W,b, q.W,b},
  // gcl[0..2]{coord_mlp1.W,b, coord_mlp2.W, edge_mlp1.W,b, edge_mlp2.W,b,
  //           node_mlp1.W,b, node_mlp2.W,b},
  // linear_in.W,b, linear_out.W,b, out_layer{same as gcl}
  struct AttP { const float *c1W,*c1b,*c2W,*kvW,*kvb,*qW,*qb; };
  struct GclP { const float *c1W,*c1b,*c2W,*e1W,*e1b,*e2W,*e2b,*n1W,*n1b,*n2W,*n2b; };
  int p = 4;
  AttP attp[3];
  for (int i = 0; i < 3; ++i) {
    attp[i].c1W=(const float*)d_in[p++]; attp[i].c1b=(const float*)d_in[p++];
    attp[i].c2W=(const float*)d_in[p++];
    attp[i].kvW=(const float*)d_in[p++]; attp[i].kvb=(const float*)d_in[p++];
    attp[i].qW =(const float*)d_in[p++]; attp[i].qb =(const float*)d_in[p++];
  }
  GclP gclp[4]; // [3] = out_layer
  for (int i = 0; i < 3; ++i) {
    GclP& g = gclp[i];
    g.c1W=(const float*)d_in[p++]; g.c1b=(const float*)d_in[p++];
    g.c2W=(const float*)d_in[p++];
    g.e1W=(const float*)d_in[p++]; g.e1b=(const float*)d_in[p++];
    g.e2W=(const float*)d_in[p++]; g.e2b=(const float*)d_in[p++];
    g.n1W=(const float*)d_in[p++]; g.n1b=(const float*)d_in[p++];
    g.n2W=(const float*)d_in[p++]; g.n2b=(const float*)d_in[p++];
  }
  const float* linW =(const float*)d_in[p++]; const float* linb =(const float*)d_in[p++];
  const float* loutW=(const float*)d_in[p++]; const float* loutb=(const float*)d_in[p++];
  { GclP& g = gclp[3];
    g.c1W=(const float*)d_in[p++]; g.c1b=(const float*)d_in[p++];
    g.c2W=(const float*)d_in[p++];
    g.e1W=(const float*)d_in[p++]; g.e1b=(const float*)d_in[p++];
    g.e2W=(const float*)d_in[p++]; g.e2b=(const float*)d_in[p++];
    g.n1W=(const float*)d_in[p++]; g.n1b=(const float*)d_in[p++];
    g.n2W=(const float*)d_in[p++]; g.n2b=(const float*)d_in[p++]; }

  // ---------------- workspace layout (floats) ----------------
  float* F = (float*)d_ws; size_t o = 0;
  float* bufA  = F+o; o += (size_t)NN*HH;
  float* bufB  = F+o; o += (size_t)NN*HH;
  float* agg   = F+o; o += (size_t)NN*HH;
  float* nscr  = F+o; o += (size_t)NN*HH;
  float* xbuf  = F+o; o += (size_t)NN*12;
  float* snode = F+o; o += (size_t)NN*12;
  float* cnt   = F+o; o += NN;
  unsigned* mbuf = (unsigned*)(F+o); o += NN;
  float* sbuf  = F+o; o += NN;
  float* sumsq = F+o; o += 16;
  float* radial= F+o; o += (size_t)ECTX*16;
  float* cdb   = F+o; o += (size_t)ECTX*12;
  float* cwb   = F+o; o += (size_t)ECTX*4;
  float* alb   = F+o; o += EATT;
  float* e0    = F+o; o += (size_t)ECTX*HH;   // also kv (EATT*256 == ECTX*128)
  float* e1    = F+o; o += (size_t)ECTX*HH;   // also q | v halves
  if (o*sizeof(float) > ws_size) return;

  auto gemm = [&](const float* A0,const int* m0,int K0,const float* A1,const int* m1,int K1,
                  const float* A2,int K2,const float* W,const float* b,const float* resid,
                  float* out,int M,int N,int flags){
    dim3 grid(M/16, N/64);
    gemm_wmma_f32<<<grid,128,0,stream>>>(A0,m0,K0,A1,m1,K1,A2,K2,W,b,resid,out,
                                         M,K0+K1+K2,N,flags);
  };

  float* h  = bufA;
  float* hN = bufB;

  auto run_gcl = [&](const GclP& P, const int* row, const int* col, int E){
    int eb = (E+255)/256;
    hipMemsetAsync(sumsq, 0, 16*sizeof(float), stream);
    k_cd_radial<<<eb,256,0,stream>>>(xbuf,row,col,cdb,radial,sumsq,E);
    k_radial_norm<<<(E*16+255)/256,256,0,stream>>>(radial,sumsq,E*16);
    // edge_mlp1: [h[row] | h[col] | radial] (272) -> 128, SiLU
    gemm(h,row,HH, h,col,HH, radial,16, P.e1W,P.e1b,nullptr, e0, E, HH, 1);
    gemm(e0,nullptr,HH, nullptr,nullptr,0, nullptr,0, P.e2W,P.e2b,nullptr, e1, E, HH, 1);
    gemm(e1,nullptr,HH, nullptr,nullptr,0, nullptr,0, P.c1W,P.c1b,nullptr, e0, E, HH, 1);
    k_matvec4<<<(E*4+255)/256,256,0,stream>>>(e0,P.c2W,cwb,E);
    hipMemsetAsync(snode, 0, (size_t)NN*12*sizeof(float), stream);
    hipMemsetAsync(cnt,   0, (size_t)NN*sizeof(float), stream);
    k_coord_scatter<<<eb,256,0,stream>>>(row,cdb,cwb,snode,cnt,E);
    k_coord_mean<<<(NN*12+255)/256,256,0,stream>>>(xbuf,snode,cnt,NN*12);
    hipMemsetAsync(agg, 0, (size_t)NN*HH*sizeof(float), stream);
    k_agg_rows<<<(E*HH+255)/256,256,0,stream>>>(row,e1,agg,E);
    // node MLPs: [h | agg] (256) -> 128 SiLU -> 128 (+h residual)
    gemm(h,nullptr,HH, agg,nullptr,HH, nullptr,0, P.n1W,P.n1b,nullptr, nscr, NN, HH, 1);
    gemm(nscr,nullptr,HH, nullptr,nullptr,0, nullptr,0, P.n2W,P.n2b, h, hN, NN, HH, 2);
    float* t = h; h = hN; hN = t;
  };

  auto run_att = [&](const AttP& P, const int* row, const int* col, int E){
    int eb = (E+255)/256;
    hipMemsetAsync(sumsq, 0, 16*sizeof(float), stream);
    k_cd_radial<<<eb,256,0,stream>>>(xbuf,row,col,cdb,radial,sumsq,E);
    k_radial_norm<<<(E*16+255)/256,256,0,stream>>>(radial,sumsq,E*16);
    float* qb = e1;                       // E x 128
    float* vb = e1 + (size_t)E*HH;        // E x 128
    gemm(h,row,HH, nullptr,nullptr,0, nullptr,0, P.qW,P.qb,nullptr, qb, E, HH, 0);
    // kv: [radial (16) | h[col] (128)] -> 256 (even cols = k, odd = v)
    gemm(radial,nullptr,16, h,col,HH, nullptr,0, P.kvW,P.kvb,nullptr, e0, E, 256, 0);
    hipMemsetAsync(mbuf, 0, (size_t)NN*sizeof(unsigned), stream);
    hipMemsetAsync(sbuf, 0, (size_t)NN*sizeof(float), stream);
    k_logit_max<<<eb,256,0,stream>>>(row,qb,e0,alb,mbuf,E);
    k_exp_sum<<<eb,256,0,stream>>>(row,alb,mbuf,sbuf,E);
    k_alpha_div<<<eb,256,0,stream>>>(row,alb,sbuf,E);
    hipMemsetAsync(agg, 0, (size_t)NN*HH*sizeof(float), stream);
    k_hagg<<<(E*HH+255)/256,256,0,stream>>>(row,alb,e0,agg,E);
    k_extract_v<<<(E*HH+255)/256,256,0,stream>>>(e0,vb,E);
    k_add<<<(NN*HH+255)/256,256,0,stream>>>(h,agg,NN*HH);   // h += segsum(alpha*v)
    gemm(vb,nullptr,HH, nullptr,nullptr,0, nullptr,0, P.c1W,P.c1b,nullptr, e1, E, HH, 1);
    k_matvec4<<<(E*4+255)/256,256,0,stream>>>(e1,P.c2W,cwb,E);
    hipMemsetAsync(snode, 0, (size_t)NN*12*sizeof(float), stream);
    k_att_coord_scatter<<<eb,256,0,stream>>>(row,cdb,cwb,alb,snode,E);
    k_add<<<(NN*12+255)/256,256,0,stream>>>(xbuf,snode,NN*12);
  };

  // ---------------- forward pass ----------------
  k_copy<<<(NN*12+255)/256,256,0,stream>>>(xbuf, x_in, NN*12);
  gemm(h_in,nullptr,INNF, nullptr,nullptr,0, nullptr,0, linW,linb,nullptr, h, NN, HH, 0);

  for (int l = 0; l < 3; ++l) {
    run_gcl(gclp[l], ctx_row, ctx_col, ECTX);
    run_att(attp[l], att_row, att_col, EATT);
  }
  run_gcl(gclp[3], ctx_row, ctx_col, ECTX);

  // outputs: h_out (N x 64) then x (N x 4 x 3), concatenated flat
  gemm(h,nullptr,HH, nullptr,nullptr,0, nullptr,0, loutW,loutb,nullptr,
       (float*)d_out, NN, ONF, 0);
  k_copy<<<(NN*12+255)/256,256,0,stream>>>((float*)d_out + (size_t)NN*ONF, xbuf, NN*12);
}